// Model_64321430225304
// MI455X (gfx1250) — compile-verified
//
#include <hip/hip_runtime.h>
#include <stdint.h>

#define B_ 64
#define L_ 512
#define T_ 2048
#define XT 8                    // columns per TDM tile
#define NTILES (T_ / XT)        // 256
#define NBUF 6                  // tile buffers resident in LDS (5 in flight)
#define RSTRIDE 9               // padded LDS row stride in floats (gcd(9,64)=1)
#define TILE_B (L_ * RSTRIDE * 4)  // bytes per tile buffer = 18432
#define NEGF (-1e15f)

typedef __attribute__((ext_vector_type(4))) unsigned int su4;
typedef __attribute__((ext_vector_type(8))) unsigned int su8;

// ---------------------------------------------------------------------------
// TDM helpers (CDNA5 async tensor path). TENSORcnt is in-order per wave, so
// wait(N-1) after issuing guarantees the oldest outstanding tile is complete.
// ---------------------------------------------------------------------------
__device__ __forceinline__ void tdm_wait_pipe() {
#if __has_builtin(__builtin_amdgcn_s_wait_tensorcnt)
  __builtin_amdgcn_s_wait_tensorcnt(NBUF - 1);
#else
  asm volatile("s_wait_tensorcnt 0x5" ::: "memory");
#endif
}
__device__ __forceinline__ void tdm_wait_0() {
#if __has_builtin(__builtin_amdgcn_s_wait_tensorcnt)
  __builtin_amdgcn_s_wait_tensorcnt(0);
#else
  asm volatile("s_wait_tensorcnt 0x0" ::: "memory");
#endif
}

// 2D TDM load: tile of XT contiguous t  x  256 rows (stride T floats) into LDS.
// Pad 1 dword after every 8 dwords -> LDS row stride 9 floats (conflict-free).
__device__ __forceinline__ void tdm_issue_half(uint64_t gaddr, unsigned lds_byte_off,
                                               unsigned rem_x) {
  su4 g0;
  g0[0] = 1u;                                         // count=1 (valid user D#)
  g0[1] = lds_byte_off;                               // lds_addr
  g0[2] = (unsigned)(gaddr & 0xFFFFFFFFull);          // global_addr[31:0]
  g0[3] = ((unsigned)(gaddr >> 32) & 0x01FFFFFFu)     // global_addr[56:32]
          | (2u << 30);                               // type = 2 ("image")
  su8 g1;
  g1[0] = (2u << 16)                                  // data_size = 4B
        | (1u << 20)                                  // pad_enable
        | (2u << 22);                                 // pad_interval: every 8 dwords (amount=1 dword)
  g1[1] = (rem_x & 0xFFFFu) << 16;                    // tensor_dim0[15:0]
  g1[2] = ((rem_x >> 16) & 0xFFFFu)                   // tensor_dim0[31:16]
        | (256u << 16);                               // tensor_dim1[15:0] = 256 rows (this half)
  g1[3] = ((unsigned)XT << 16);                       // tensor_dim1 hi=0, tile_dim0 = 8
  g1[4] = 256u;                                       // tile_dim1 = 256, tile_dim2 = 0
  g1[5] = (unsigned)T_;                               // tensor_dim0_stride[31:0] = T
  g1[6] = 0u;
  g1[7] = 0u;
  asm volatile("tensor_load_to_lds %0, %1" :: "s"(g0), "s"(g1) : "memory");
}

// ---------------------------------------------------------------------------
// Forward DP: one workgroup per batch b, 512 threads (16 wave32s), thread = row.
// Emits only the 1-bit decision mask D[b][t][w] (bit l%32 of word w=l/32):
//   bit(l) = ( beta_t[(l-1) mod L] > beta_t[l] )
// which is exactly is_go for the reference backtrace (incl. numpy wrap of -1).
// Waves 0 and 8 each drive a TDM engine for half the rows, NBUF-deep pipeline.
// ---------------------------------------------------------------------------
struct SharedFwd {
  float tile[NBUF][L_ * RSTRIDE];  // 110592 B of TDM destinations
  float beta[2][L_];               // 4096 B double-buffered DP column
};

__global__ __launch_bounds__(L_) void viterbi_forward_tdm(
    const float* __restrict__ lp, unsigned* __restrict__ D) {
  __shared__ SharedFwd s;
  const int b    = blockIdx.x;
  const int l    = threadIdx.x;
  const int wave = l >> 5;
  const int lane = l & 31;

  const bool issuer = ((wave & 7) == 0);            // waves 0 and 8
  const unsigned half = (wave >> 3) & 1;            // rows [0,256) / [256,512)
  const uint64_t base = (uint64_t)(uintptr_t)lp +
                        4ull * ((uint64_t)b * L_ * T_ + (uint64_t)half * 256u * T_);
  const unsigned ldsHalf = half * 256u * RSTRIDE * 4u;

  // prologue: fill the pipeline with tiles 0..NBUF-2
  if (issuer) {
    for (int k = 0; k < NBUF - 1; ++k)
      tdm_issue_half(base + 4ull * (uint64_t)(XT * k),
                     (unsigned)((k % NBUF) * TILE_B) + ldsHalf,
                     (unsigned)(T_ - XT * k));
  }

  float beta = NEGF;
  for (int k = 0; k < NTILES; ++k) {
    if (issuer) {
      const int ka = k + NBUF - 1;                  // issue ahead
      if (ka < NTILES)
        tdm_issue_half(base + 4ull * (uint64_t)(XT * ka),
                       (unsigned)((ka % NBUF) * TILE_B) + ldsHalf,
                       (unsigned)(T_ - XT * ka));
      if (k <= NTILES - NBUF) tdm_wait_pipe();      // oldest (tile k) complete
      else                    tdm_wait_0();         // drain at the tail
    }
    __syncthreads();  // tile k visible to all waves

    const float* tile = s.tile[k % NBUF];
#pragma unroll
    for (int j = 0; j < XT; ++j) {
      const int c   = XT * k + j;
      const float v = tile[l * RSTRIDE + j];
      float nw;
      if (c == 0) {
        nw = (l == 0) ? v : NEGF;  // beta0 init column
      } else {
        const float shifted = (l == 0) ? NEGF : s.beta[(c - 1) & 1][l - 1];
        nw = fmaxf(beta, shifted) + v;
      }
      s.beta[c & 1][l] = nw;
      __syncthreads();  // single barrier per column (double-buffered beta)
      const float nbr = s.beta[c & 1][(l + L_ - 1) & (L_ - 1)];
      const bool pred = nbr > nw;
#if __has_builtin(__builtin_amdgcn_ballot_w32)
      const unsigned mask = __builtin_amdgcn_ballot_w32(pred);
#else
      const unsigned mask = (unsigned)__ballot(pred);
#endif
      if (lane == 0) D[((size_t)b * T_ + c) * 16 + wave] = mask;
      beta = nw;
    }
  }
}

// ---------------------------------------------------------------------------
// Backtrace: one block per batch. Stage the 128KB bit-slice into LDS
// (CDNA5: 320KB LDS/WGP) so the 2047-long dependent chain runs at LDS latency.
// All-integer walk (values are exact small integers); floats only at the store.
// ---------------------------------------------------------------------------
__global__ __launch_bounds__(256) void viterbi_backtrace(
    const unsigned* __restrict__ D, const int* __restrict__ tlen,
    const int* __restrict__ mlen, float* __restrict__ out) {
  __shared__ unsigned ldsD[T_ * 16];  // 131072 bytes
  const int b = blockIdx.x;
  const uint4* Db4 = (const uint4*)(D + (size_t)b * T_ * 16);
  uint4* ldsD4 = (uint4*)ldsD;
  for (int i = threadIdx.x; i < T_ * 4; i += 256) ldsD4[i] = Db4[i];
  __syncthreads();

  if (threadIdx.x == 0) {
    int ri = tlen[b] - 1;
    int ci = mlen[b] - 1;
    float* ob = out + (size_t)b * T_;
    ob[T_ - 1] = (float)ri;  // path = [r0, steps...] reversed
    for (int st = 1; st < T_; ++st) {
      int tc = ci - 1;                       // >= -2; numpy wrap:
      if (tc < 0) tc += T_;
      const int row = ri & (L_ - 1);         // ri in [-1, L-1]; -1 -> L-1
      const unsigned w = ldsD[tc * 16 + (row >> 5)];
      const int is_go = (int)((w >> (row & 31)) & 1u);
      ri = max(ri - is_go + 1, 0) - 1;       // relu(r - is_go + 1) - 1
      ci = max(ci, 0) - 1;                   // relu(c) - 1
      ob[T_ - 1 - st] = (float)ri;
    }
  }
}

// ---------------------------------------------------------------------------
extern "C" void kernel_launch(void* const* d_in, const int* in_sizes, int n_in,
                              void* d_out, int out_size, void* d_ws, size_t ws_size,
                              hipStream_t stream) {
  (void)in_sizes; (void)n_in; (void)out_size; (void)ws_size;
  const float* lp   = (const float*)d_in[0];
  const int*   tlen = (const int*)d_in[1];
  const int*   mlen = (const int*)d_in[2];
  float*       out  = (float*)d_out;
  unsigned*    D    = (unsigned*)d_ws;  // 64*2048*16 u32 = 8 MB decision bitmask

  viterbi_forward_tdm<<<dim3(B_), dim3(L_), 0, stream>>>(lp, D);
  viterbi_backtrace<<<dim3(B_), dim3(256), 0, stream>>>(D, tlen, mlen, out);
}